// EntNet_670014898558
// MI455X (gfx1250) — compile-verified
//
#include <hip/hip_runtime.h>

// Problem constants (from reference): B=64, S=128, L=32, D=100, M=20, V=50000
#define Bb   64
#define Ss   128
#define Ll   32
#define Dd   100
#define Mm   20
#define DP   112   // D padded to 7 WMMA N-tiles of 16
#define MP   32    // M padded to 2 WMMA M-tiles of 16
#define NTHR 256   // 8 waves (wave32)
#define NWAVE 8
#define NTILE 14   // 2 M-tiles x 7 N-tiles; each wave owns tile 'wave' (+ 'wave+8' for waves 0..5)
#define KSTEPS 25  // K = 100 = 25 exact k-steps of 4

typedef __attribute__((ext_vector_type(2))) float v2f;
typedef __attribute__((ext_vector_type(8))) float v8f;

// ---------------------------------------------------------------------------
// Kernel 1: sents[b,s,d] = sum_l emb[tokens[b,s,l], d] * mult[l, d]
// Coalesced over d; tokens broadcast per block. emb (20MB) stays L2-resident.
// ---------------------------------------------------------------------------
__global__ __launch_bounds__(128) void sents_kernel(
    const int* __restrict__ tokens, const float* __restrict__ emb,
    const float* __restrict__ mult, float* __restrict__ sents)
{
    const int bs = blockIdx.x;         // 0 .. B*S-1
    const int d  = threadIdx.x;
    if (d >= Dd) return;
    const int* trow = tokens + bs * Ll;
    float acc = 0.f;
    #pragma unroll 4
    for (int l = 0; l < Ll; ++l) {
        const int tok = trow[l];
        acc += emb[tok * Dd + d] * mult[l * Dd + d];
    }
    sents[bs * Dd + d] = acc;
}

// ---------------------------------------------------------------------------
// Kernel 2: loop-invariant Vk[m,d] = sum_k keys[m,k] * Vw[d,k]
// ---------------------------------------------------------------------------
__global__ __launch_bounds__(128) void vk_kernel(
    const float* __restrict__ keys, const float* __restrict__ Vw,
    float* __restrict__ VkBuf)
{
    const int m = blockIdx.x;
    const int d = threadIdx.x;
    if (d >= Dd) return;
    float acc = 0.f;
    for (int k = 0; k < Dd; ++k) acc += keys[m * Dd + k] * Vw[d * Dd + k];
    VkBuf[m * Dd + d] = acc;
}

// ---------------------------------------------------------------------------
// Kernel 3: persistent recurrent scan. One block per batch element.
// State mem[b] (padded 32x112, pad rows/cols kept zero) lives in LDS for all
// 128 steps. Per-step GEMM mem @ Uw^T via V_WMMA_F32_16X16X4_F32.
// B-fragments (Uw^T slabs) are scan-invariant -> hoisted into VGPRs once, so
// the steady-state inner loop is ds_load_b64(A) + v_wmma per k-step.
// ---------------------------------------------------------------------------
__global__ __launch_bounds__(NTHR) void entnet_scan_kernel(
    const float* __restrict__ sents, const float* __restrict__ keys,
    const float* __restrict__ Uw,    const float* __restrict__ Ww,
    const float* __restrict__ VkBuf, const float* __restrict__ prelu_a,
    float* __restrict__ out)
{
    const int b    = blockIdx.x;
    const int tid  = threadIdx.x;
    const int wave = tid >> 5;
    const int lane = tid & 31;
    const float pa = prelu_a[0];

    __shared__ float Uwt  [Dd * DP];  // Uw^T: [k][d], d padded with zeros (44.8 KB)
    __shared__ float memS [MP * DP];  // memory state, pad rows/cols zero (14 KB)
    __shared__ float sS   [DP];       // current sentence vector (padded)
    __shared__ float WsS  [DP];       // Ww @ s (padded)
    __shared__ float gateS[Mm];       // sigmoid gate * null-sentence mask
    __shared__ float scaleS[Mm];      // 1/||mem_slot||

    // ---- setup: zero padded state, stage Uw^T, init mem = keys ----
    for (int i = tid; i < MP * DP; i += NTHR) memS[i] = 0.f;
    for (int i = tid; i < Dd * DP; i += NTHR) {
        const int k = i / DP, d = i - k * DP;
        Uwt[i] = (d < Dd) ? Uw[d * Dd + k] : 0.f;   // Uwt[k][d] = Uw[d][k]
    }
    __syncthreads();
    for (int i = tid; i < Mm * Dd; i += NTHR) {
        const int m = i / Dd, d = i - m * Dd;
        memS[m * DP + d] = keys[i];
    }
    __syncthreads();

    const int r15 = lane & 15;        // tile row/col within half-wave
    const int kh  = (lane >> 4) * 2;  // K sub-offset for A/B fragments
    const int mh  = (lane >> 4) * 8;  // M sub-offset for C/D fragment

    // static tile assignment: tile0 = wave (all 8 waves), tile1 = wave+8 (waves 0..5)
    const int  t0   = wave;
    const int  t1   = wave + NWAVE;
    const bool has1 = (t1 < NTILE);                 // wave-uniform
    const int  mb0 = (t0 / 7) * 16, nb0 = (t0 % 7) * 16;
    const int  mb1 = (t1 / 7) * 16, nb1 = (t1 % 7) * 16;

    // ---- hoist scan-invariant B-fragments (Uw^T) into registers ----
    // B 4x16 f32 (mirrored A layout): lanes 0-15 K={kk,kk+1}, lanes 16-31 K={kk+2,kk+3}
    v2f bf0[KSTEPS], bf1[KSTEPS];
    #pragma unroll
    for (int j = 0; j < KSTEPS; ++j) {
        const int kk = j * 4;
        bf0[j].x = Uwt[(kk + kh)     * DP + nb0 + r15];
        bf0[j].y = Uwt[(kk + kh + 1) * DP + nb0 + r15];
    }
    if (has1) {
        #pragma unroll
        for (int j = 0; j < KSTEPS; ++j) {
            const int kk = j * 4;
            bf1[j].x = Uwt[(kk + kh)     * DP + nb1 + r15];
            bf1[j].y = Uwt[(kk + kh + 1) * DP + nb1 + r15];
        }
    } else {
        #pragma unroll
        for (int j = 0; j < KSTEPS; ++j) bf1[j] = v2f{0.f, 0.f};
    }

    // 16x16x100 f32 WMMA tile; A streamed from LDS, B from registers
    auto gemm_tile = [&](int mb, const v2f* bf) -> v8f {
        v8f c = {};
        #pragma unroll
        for (int j = 0; j < KSTEPS; ++j) {
            const int kk = j * 4;
            // A 16x4 f32: lanes 0-15 K={kk,kk+1}, lanes 16-31 K={kk+2,kk+3}
            const v2f a = *(const v2f*)&memS[(mb + r15) * DP + kk + kh];
            c = __builtin_amdgcn_wmma_f32_16x16x4_f32(
                    /*neg_a=*/false, a, /*neg_b=*/false, bf[j],
                    /*c_mod=*/(short)0, c, /*reuse_a=*/false, /*reuse_b=*/false);
        }
        return c;
    };

    // cand = PReLU(gemm + Vk + Ws); mem += cand * (gate*mask)
    auto apply_tile = [&](int mb, int nb, v8f c) {
        #pragma unroll
        for (int r = 0; r < 8; ++r) {
            const int m = mb + mh + r;      // C/D layout: M = r + 8*(lane>=16)
            const int n = nb + r15;
            if (m < Mm && n < Dd) {
                float x = c[r] + VkBuf[m * Dd + n] + WsS[n];
                x = (x >= 0.f) ? x : pa * x;
                memS[m * DP + n] += x * gateS[m];
            }
        }
    };

    for (int t = 0; t < Ss; ++t) {
        // ---- load sentence vector (zero-padded to 112) ----
        if (tid < DP) sS[tid] = (tid < Dd) ? sents[(b * Ss + t) * Dd + tid] : 0.f;
        // hide next step's fetch latency behind this step's GEMM
        if (t + 1 < Ss && tid < 4)
            __builtin_prefetch(&sents[(b * Ss + t + 1) * Dd + tid * 32], 0, 3);
        __syncthreads();

        // ---- gates (threads 0..19) and Ws = s @ Ww^T (threads 32..143), concurrent ----
        if (tid < Mm) {
            float mg = 0.f, kg = 0.f;
            #pragma unroll 4
            for (int k = 0; k < Dd; ++k) {
                const float sv = sS[k];
                mg += sv * memS[tid * DP + k];
                kg += sv * keys[tid * Dd + k];
            }
            const float z = mg + kg;
            const float g = 1.f / (1.f + expf(-z));
            // null-sentence mask: sigmoid==0.5 (i.e. z==0) -> gated contribution 0
            gateS[tid] = (z == 0.f) ? 0.f : g;
        } else if (tid >= 32 && tid < 32 + DP) {
            const int d = tid - 32;
            float w = 0.f;
            if (d < Dd) {
                #pragma unroll 4
                for (int k = 0; k < Dd; ++k) w += sS[k] * Ww[d * Dd + k];
            }
            WsS[d] = w;
        }
        __syncthreads();

        // ---- cand_pre = mem @ Uw^T via f32 WMMA; fixed-register accumulators ----
        v8f acc0 = gemm_tile(mb0, bf0);
        v8f acc1 = {};
        if (has1) acc1 = gemm_tile(mb1, bf1);
        __syncthreads();   // all memS reads complete before anyone updates it

        apply_tile(mb0, nb0, acc0);
        if (has1) apply_tile(mb1, nb1, acc1);
        __syncthreads();

        // ---- per-slot L2 normalization ----
        if (tid < Mm) {
            float ss = 0.f;
            #pragma unroll 4
            for (int d = 0; d < Dd; ++d) {
                const float v = memS[tid * DP + d];
                ss += v * v;
            }
            scaleS[tid] = 1.f / sqrtf(ss);
        }
        __syncthreads();
        for (int i = tid; i < Mm * DP; i += NTHR) {
            const int m = i / DP;
            memS[i] *= scaleS[m];   // padded cols are zero and stay zero
        }
        __syncthreads();
    }

    // ---- write out mem[b] ----
    for (int i = tid; i < Mm * Dd; i += NTHR) {
        const int m = i / Dd, d = i - m * Dd;
        out[b * Mm * Dd + i] = memS[m * DP + d];
    }
}

// ---------------------------------------------------------------------------
// Host-side launch. Inputs (setup_inputs order):
//   0 tokens[B,S,L] i32, 1 emb[V,D] f32, 2 keys[M,D] f32, 3 mult[L,D] f32,
//   4 Uw[D,D], 5 Vw[D,D], 6 Ww[D,D], 7 prelu_a[1]
// Output: mem [B,M,D] f32.  Workspace: sents (B*S*D) + Vk (M*D) floats.
// ---------------------------------------------------------------------------
extern "C" void kernel_launch(void* const* d_in, const int* in_sizes, int n_in,
                              void* d_out, int out_size, void* d_ws, size_t ws_size,
                              hipStream_t stream) {
    (void)in_sizes; (void)n_in; (void)out_size; (void)ws_size;
    const int*   tokens  = (const int*)  d_in[0];
    const float* emb     = (const float*)d_in[1];
    const float* keys    = (const float*)d_in[2];
    const float* mult    = (const float*)d_in[3];
    const float* Uw      = (const float*)d_in[4];
    const float* Vw      = (const float*)d_in[5];
    const float* Ww      = (const float*)d_in[6];
    const float* prelu_a = (const float*)d_in[7];
    float* out = (float*)d_out;

    float* sents = (float*)d_ws;                  // B*S*D floats (3.28 MB)
    float* VkBuf = sents + (size_t)Bb * Ss * Dd;  // M*D floats

    sents_kernel<<<Bb * Ss, 128, 0, stream>>>(tokens, emb, mult, sents);
    vk_kernel<<<Mm, 128, 0, stream>>>(keys, Vw, VkBuf);
    entnet_scan_kernel<<<Bb, NTHR, 0, stream>>>(sents, keys, Uw, Ww, VkBuf,
                                                prelu_a, out);
}